// TopoLossMSE2D_62088047231145
// MI455X (gfx1250) — compile-verified
//
#include <hip/hip_runtime.h>
#include <stdint.h>

#define HW    512
#define CH    48
#define CPL   (HW * HW)          // elements per channel plane
#define BIGV  1.0e10f            // FastGeodis 'v' (note: BIGV + 1.41f == BIGV in f32)
#define SQ2   1.41421356237309515f

// ---- CDNA5 async-to-LDS availability probes ------------------------------
#if defined(__has_builtin)
#if __has_builtin(__builtin_amdgcn_global_load_async_to_lds_b128)
#define HAVE_ASYNC_LDS 1
#endif
#if __has_builtin(__builtin_amdgcn_s_wait_asynccnt)
#define HAVE_WAIT_ASYNC_BUILTIN 1
#endif
#endif

typedef float v8f __attribute__((ext_vector_type(8)));
typedef float v2f __attribute__((ext_vector_type(2)));
typedef int   v4i __attribute__((vector_size(16)));   // matches builtin's param type

__device__ __forceinline__ float getc(const float4& v, int i) {
  return i == 0 ? v.x : i == 1 ? v.y : i == 2 ? v.z : v.w;
}
__device__ __forceinline__ void setc(float4& v, int i, float x) {
  if (i == 0) v.x = x; else if (i == 1) v.y = x; else if (i == 2) v.z = x; else v.w = x;
}

// ---- Init: D = V*(1-seed). mode 0: seed = (pred >= 0.4) ; mode 1: seed = target
__global__ void k_init(const float* __restrict__ src, float* __restrict__ D,
                       int mode, int n) {
  int i = blockIdx.x * blockDim.x + threadIdx.x;
  int stride = gridDim.x * blockDim.x;
  for (; i < n; i += stride) {
    float s = src[i];
    D[i] = mode ? (BIGV * (1.0f - s)) : ((s >= 0.4f) ? 0.0f : BIGV);
  }
}

// ---- Vertical raster pass: one wave32 per channel, lane owns 16 columns.
// dir=+1: top-down, dir=-1: bottom-up. prev row lives in registers; the
// 3-neighborhood across lane boundaries needs only 2 intra-wave shuffles.
__global__ __launch_bounds__(32) void k_vpass(float* __restrict__ D, int dir) {
  float* base = D + (size_t)blockIdx.x * CPL;
  const int lane = threadIdx.x;
  float* colp = base + lane * 16;

  float p[16];
#pragma unroll
  for (int j = 0; j < 16; ++j) p[j] = BIGV;   // V+1==V in f32 -> first row is a no-op

  int r = (dir > 0) ? 0 : (HW - 1);
  for (int i = 0; i < HW; ++i) {
    float* rowp = colp + (size_t)r * HW;
    float4 q0 = ((const float4*)rowp)[0];
    float4 q1 = ((const float4*)rowp)[1];
    float4 q2 = ((const float4*)rowp)[2];
    float4 q3 = ((const float4*)rowp)[3];
    if (i + 1 < HW) __builtin_prefetch(rowp + dir * HW, 0, 0);  // global_prefetch_b8

    float cur[16] = {q0.x, q0.y, q0.z, q0.w, q1.x, q1.y, q1.z, q1.w,
                     q2.x, q2.y, q2.z, q2.w, q3.x, q3.y, q3.z, q3.w};

    float left  = __shfl_up(p[15], 1, 32);
    float right = __shfl_down(p[0], 1, 32);
    if (lane == 0)  left  = BIGV;
    if (lane == 31) right = BIGV;

    float np[16];
#pragma unroll
    for (int j = 0; j < 16; ++j) {
      float a = ((j == 0)  ? left  : p[j - 1]) + SQ2;
      float b = p[j] + 1.0f;
      float c = ((j == 15) ? right : p[j + 1]) + SQ2;
      np[j] = fminf(cur[j], fminf(a, fminf(b, c)));
    }
    float4 o0 = {np[0],  np[1],  np[2],  np[3]};
    float4 o1 = {np[4],  np[5],  np[6],  np[7]};
    float4 o2 = {np[8],  np[9],  np[10], np[11]};
    float4 o3 = {np[12], np[13], np[14], np[15]};
    ((float4*)rowp)[0] = o0; ((float4*)rowp)[1] = o1;
    ((float4*)rowp)[2] = o2; ((float4*)rowp)[3] = o3;
#pragma unroll
    for (int j = 0; j < 16; ++j) p[j] = np[j];
    r += dir;
  }
}

// ---- Horizontal raster pass: lane owns 16 rows; scan over columns with
// 4-column float4 register batching (amortizes the strided column traffic;
// the 48MB stack is L2-resident anyway). dir=+1: L->R, dir=-1: R->L.
__global__ __launch_bounds__(32) void k_hpass(float* __restrict__ D, int dir) {
  float* base = D + (size_t)blockIdx.x * CPL;
  const int lane = threadIdx.x;
  const int j0 = lane * 16;

  float p[16];
#pragma unroll
  for (int j = 0; j < 16; ++j) p[j] = BIGV;   // first column becomes a no-op

  for (int b = 0; b < HW / 4; ++b) {
    const int xb = (dir > 0) ? (b * 4) : (HW - 4 - b * 4);
    float4 f[16];
#pragma unroll
    for (int j = 0; j < 16; ++j)
      f[j] = *(const float4*)(base + (size_t)(j0 + j) * HW + xb);
    if (b + 1 < HW / 4)
      __builtin_prefetch(base + (size_t)j0 * HW + (xb + 4 * dir), 0, 0);

#pragma unroll
    for (int k = 0; k < 4; ++k) {
      const int ci = (dir > 0) ? k : (3 - k);
      float up = __shfl_up(p[15], 1, 32);
      float dn = __shfl_down(p[0], 1, 32);
      if (lane == 0)  up = BIGV;
      if (lane == 31) dn = BIGV;
      float np[16];
#pragma unroll
      for (int j = 0; j < 16; ++j) {
        float a = ((j == 0)  ? up : p[j - 1]) + SQ2;
        float m = p[j] + 1.0f;
        float c = ((j == 15) ? dn : p[j + 1]) + SQ2;
        np[j] = fminf(getc(f[j], ci), fminf(a, fminf(m, c)));
      }
#pragma unroll
      for (int j = 0; j < 16; ++j) { p[j] = np[j]; setc(f[j], ci, np[j]); }
    }
#pragma unroll
    for (int j = 0; j < 16; ++j)
      *(float4*)(base + (size_t)(j0 + j) * HW + xb) = f[j];
  }
}

// ---- Window statistics: one block per (channel, 64x64 window); stage the
// 16KB tile into LDS via async loads (ASYNCcnt path), then min/max reduce.
__global__ __launch_bounds__(256) void k_winstat(const float* __restrict__ D,
                                                 float* __restrict__ wmin,
                                                 float* __restrict__ wmax) {
  __shared__ float tile[64 * 64];
  const int w  = blockIdx.x & 63;
  const int ch = blockIdx.x >> 6;
  const int wy = w >> 3, wx = w & 7;
  const float* src = D + (size_t)ch * CPL + (size_t)(wy * 64) * HW + wx * 64;
  const int t = threadIdx.x;

#pragma unroll
  for (int k = 0; k < 4; ++k) {
    const int e = (k * 256 + t) * 4;             // 16B-aligned element index
    const int row = e >> 6, col = e & 63;
    const float* gp = src + (size_t)row * HW + col;
#if HAVE_ASYNC_LDS
    __builtin_amdgcn_global_load_async_to_lds_b128(
        (__attribute__((address_space(1))) v4i*)(uintptr_t)gp,
        (__attribute__((address_space(3))) v4i*)(uint32_t)(uintptr_t)&tile[e],
        0, 0);
#else
    *(float4*)&tile[e] = *(const float4*)gp;
#endif
  }
#if HAVE_ASYNC_LDS
#if HAVE_WAIT_ASYNC_BUILTIN
  __builtin_amdgcn_s_wait_asynccnt(0);
#else
  asm volatile("s_wait_asynccnt 0" ::: "memory");
#endif
#endif
  __syncthreads();

  float mn = 3.4e38f, mx = -3.4e38f;
#pragma unroll
  for (int i = 0; i < 16; ++i) {
    float v = tile[t + i * 256];
    mn = fminf(mn, v); mx = fmaxf(mx, v);
  }
#pragma unroll
  for (int o = 16; o > 0; o >>= 1) {
    mn = fminf(mn, __shfl_xor(mn, o, 32));
    mx = fmaxf(mx, __shfl_xor(mx, o, 32));
  }
  __shared__ float rmn[8], rmx[8];
  if ((t & 31) == 0) { rmn[t >> 5] = mn; rmx[t >> 5] = mx; }
  __syncthreads();
  if (t == 0) {
    float a = rmn[0], b = rmx[0];
    for (int i = 1; i < 8; ++i) { a = fminf(a, rmn[i]); b = fmaxf(b, rmx[i]); }
    wmin[blockIdx.x] = a;
    wmax[blockIdx.x] = b;
  }
}

// ---- Final: normalize window stats per channel, sum 6144 squared diffs via
// v_wmma_f32_16x16x4_f32 (D = A x ones + C; total = sum(D)/16), write outputs.
__global__ __launch_bounds__(32) void k_final(const float* __restrict__ wmin,
                                              const float* __restrict__ wmax,
                                              float* __restrict__ out) {
  __shared__ float smn[96], smx[96];
  const int lane = threadIdx.x;
  for (int c = lane; c < 96; c += 32) {
    float mn = 3.4e38f, mx = -3.4e38f;
    for (int w = 0; w < 64; ++w) {
      mn = fminf(mn, wmin[c * 64 + w]);
      mx = fmaxf(mx, wmax[c * 64 + w]);
    }
    smn[c] = mn; smx[c] = mx;
  }
  __syncthreads();

  v8f acc = {0.f, 0.f, 0.f, 0.f, 0.f, 0.f, 0.f, 0.f};
  const v2f ones = {1.0f, 1.0f};
  for (int s = 0; s < 96; ++s) {
    v2f a;
#pragma unroll
    for (int vi = 0; vi < 2; ++vi) {
      const int g = s * 64 + vi * 32 + lane;    // 0..6143, each term exactly once
      const int c = g >> 7;                     // channel pair 0..47
      const int r = g & 127;
      const int w = r & 63;
      const float* arr = (r >= 64) ? wmax : wmin;
      const float pv = arr[c * 64 + w];
      const float tv = arr[(48 + c) * 64 + w];
      const float pn = smn[c];
      const float ps = 1.0f / (smx[c] - pn + 1e-6f);
      const float tn = smn[48 + c];
      const float ts = 1.0f / (smx[48 + c] - tn + 1e-6f);
      const float d = (pv - pn) * ps - (tv - tn) * ts;
      a[vi] = d * d;
    }
    acc = __builtin_amdgcn_wmma_f32_16x16x4_f32(
        false, a, false, ones, (short)0, acc, false, false);
  }
  float ssum = 0.0f;
#pragma unroll
  for (int i = 0; i < 8; ++i) ssum += acc[i];
#pragma unroll
  for (int o = 16; o > 0; o >>= 1) ssum += __shfl_xor(ssum, o, 32);
  const float total     = ssum * (1.0f / 16.0f);   // each row-sum replicated 16x in D
  const float mean_loss = total * (1.0f / 48.0f);  // mean over B*C
  if (lane == 0) {
    out[0] = mean_loss * 0.005f;  // final_loss
    out[1] = mean_loss;           // mean_loss
  }
}

extern "C" void kernel_launch(void* const* d_in, const int* in_sizes, int n_in,
                              void* d_out, int out_size, void* d_ws, size_t ws_size,
                              hipStream_t stream) {
  const float* pred   = (const float*)d_in[0];
  const float* target = (const float*)d_in[1];
  float* D    = (float*)d_ws;                  // 48 * 512 * 512 floats (48 MiB)
  float* wmin = D + (size_t)CH * CPL;          // 96 * 64 floats
  float* wmax = wmin + 96 * 64;                // 96 * 64 floats
  float* outp = (float*)d_out;
  const int n = CH * CPL;

  for (int phase = 0; phase < 2; ++phase) {    // 0: pred (binarized), 1: target
    k_init<<<2048, 256, 0, stream>>>(phase ? target : pred, D, phase, n);
    for (int it = 0; it < 2; ++it) {
      k_vpass<<<CH, 32, 0, stream>>>(D, +1);   // top-down
      k_vpass<<<CH, 32, 0, stream>>>(D, -1);   // bottom-up
      k_hpass<<<CH, 32, 0, stream>>>(D, +1);   // left-right
      k_hpass<<<CH, 32, 0, stream>>>(D, -1);   // right-left
    }
    k_winstat<<<CH * 64, 256, 0, stream>>>(D, wmin + phase * CH * 64,
                                              wmax + phase * CH * 64);
  }
  k_final<<<1, 32, 0, stream>>>(wmin, wmax, outp);
}